// FAWMF_31147102830631
// MI455X (gfx1250) — compile-verified
//
#include <hip/hip_runtime.h>
#include <hip/hip_bf16.h>

#define NUM_USERS   100000
#define NUM_ITEMS   50000
#define NUM_COMM    32
#define EMBED_DIM   64
#define NI          2000000
#define BATCH       16384
#define WEIGHT_DECAY 1e-4f
#define EPS_C       0.001f
#define LOG_CLAMP   -100.0f

typedef __attribute__((ext_vector_type(2))) float v2f;
typedef __attribute__((ext_vector_type(8))) float v8f;

__device__ __forceinline__ float wave_sum(float v) {
  #pragma unroll
  for (int off = 16; off > 0; off >>= 1) v += __shfl_xor(v, off, 32);
  return v;
}

__device__ __forceinline__ float wave_max(float v) {
  #pragma unroll
  for (int off = 16; off > 0; off >>= 1) v = fmaxf(v, __shfl_xor(v, off, 32));
  return v;
}

// ---------------- zero workspace regions ----------------
__global__ void k_zero(float* __restrict__ p, int n) {
  int i = blockIdx.x * blockDim.x + threadIdx.x;
  int stride = gridDim.x * blockDim.x;
  for (; i < n; i += stride) p[i] = 0.0f;
}

// ---------------- softmax(theta_user) rows + reg3 partial ----------------
__global__ void k_softmax(const float* __restrict__ theta_u,
                          float* __restrict__ theta, float* __restrict__ acc) {
  int lane   = threadIdx.x & 31;
  int wave   = (blockIdx.x * blockDim.x + threadIdx.x) >> 5;
  int nwaves = (gridDim.x * blockDim.x) >> 5;
  float sq = 0.0f;
  for (int r = wave; r < NUM_USERS; r += nwaves) {
    float x = theta_u[r * NUM_COMM + lane];
    sq += x * x;
    float m = wave_max(x);
    float e = expf(x - m);
    float s = wave_sum(e);
    theta[r * NUM_COMM + lane] = e / s;
  }
  sq = wave_sum(sq);
  if (lane == 0) atomicAdd(&acc[2], sq);
}

// ---------------- SpMM scatter: z[item,:] += val * theta[user,:] ----------------
// Only the second half of the symmetric edge list contributes to item rows
// (first half gathers zero-padded item theta in the reference).
__global__ void k_spmm(const int* __restrict__ erows, const int* __restrict__ ecols,
                       const float* __restrict__ evals, const float* __restrict__ theta,
                       float* __restrict__ z) {
  int lane   = threadIdx.x & 31;
  int wave   = (blockIdx.x * blockDim.x + threadIdx.x) >> 5;
  int nwaves = (gridDim.x * blockDim.x) >> 5;
  for (int e = wave; e < NI; e += nwaves) {
    int   item = erows[NI + e] - NUM_USERS;
    int   user = ecols[NI + e];
    float v    = evals[NI + e];
    atomicAdd(&z[(size_t)item * NUM_COMM + lane],
              v * theta[(size_t)user * NUM_COMM + lane]);
  }
}

// ---------------- z1 = sigmoid(z*w1 + w2) in place, + reg2 partial ----------------
__global__ void k_z1(float* __restrict__ z, const float* __restrict__ w1,
                     const float* __restrict__ w2, float* __restrict__ acc) {
  int idx = blockIdx.x * blockDim.x + threadIdx.x;
  int stride = gridDim.x * blockDim.x;
  float sq = 0.0f;
  for (int t = idx; t < NUM_ITEMS * NUM_COMM; t += stride) {
    int i = t >> 5;
    float a = w1[i], b = w2[i];
    if ((t & 31) == 0) sq += a * a + b * b;   // count each item once
    float x = z[t] * a + b;
    z[t] = 1.0f / (1.0f + expf(-x));
  }
  sq = wave_sum(sq);
  if ((threadIdx.x & 31) == 0) atomicAdd(&acc[1], sq);
}

// ---------------- WMMA rowwise dots: pos/neg scores over D=64, + reg1 ----------------
// One wave = one 16-row batch tile. diag(A x B^T) via V_WMMA_F32_16X16X4_F32,
// accumulated over 16 K-steps of 4. A and B lane layouts coincide:
// lane l loads row (l&15), k-pair 4*s + 2*(l>>4).
__global__ void k_scores(const float* __restrict__ ue_tab, const float* __restrict__ ie_tab,
                         const int* __restrict__ users, const int* __restrict__ pos,
                         const int* __restrict__ neg,
                         float* __restrict__ pos_s, float* __restrict__ neg_s,
                         float* __restrict__ acc) {
  int lane = threadIdx.x & 31;
  int tile = (blockIdx.x * blockDim.x + threadIdx.x) >> 5;   // exactly BATCH/16 waves
  int m = lane & 15;
  int h = lane >> 4;
  int row = tile * 16 + m;
  const float* uep = ue_tab + (size_t)users[row] * EMBED_DIM;
  const float* pep = ie_tab + (size_t)pos[row]   * EMBED_DIM;
  const float* nep = ie_tab + (size_t)neg[row]   * EMBED_DIM;

  v8f cp = {}; v8f cn = {};
  float sq = 0.0f;
  #pragma unroll
  for (int s = 0; s < 16; ++s) {
    int k = 4 * s + 2 * h;
    v2f a  = *(const v2f*)(uep + k);
    v2f bp = *(const v2f*)(pep + k);
    v2f bn = *(const v2f*)(nep + k);
    sq += a[0]*a[0] + a[1]*a[1] + bp[0]*bp[0] + bp[1]*bp[1] + bn[0]*bn[0] + bn[1]*bn[1];
    cp = __builtin_amdgcn_wmma_f32_16x16x4_f32(false, a, false, bp, (short)0, cp, false, false);
    cn = __builtin_amdgcn_wmma_f32_16x16x4_f32(false, a, false, bn, (short)0, cn, false, false);
  }
  sq = wave_sum(sq);
  if (lane == 0) atomicAdd(&acc[0], sq);

  int base = tile * 16;
  #pragma unroll
  for (int i = 0; i < 8; ++i) {
    int mm = i + ((lane >= 16) ? 8 : 0);       // row owned by VGPR i in this half
    if ((lane & 15) == mm) {                   // diagonal element D[mm][mm]
      pos_s[base + mm] = cp[i];
      neg_s[base + mm] = cn[i];
    }
  }
}

// ---------------- WMMA rowwise dots: gamma = theta[u] . z1[i] over C=32 ----------------
__global__ void k_gamma(const float* __restrict__ theta, const float* __restrict__ z1,
                        const int* __restrict__ users, const int* __restrict__ pos,
                        const int* __restrict__ neg,
                        float* __restrict__ gpos, float* __restrict__ gneg) {
  int lane = threadIdx.x & 31;
  int tile = (blockIdx.x * blockDim.x + threadIdx.x) >> 5;
  int m = lane & 15;
  int h = lane >> 4;
  int row = tile * 16 + m;
  const float* tp = theta + (size_t)users[row] * NUM_COMM;
  const float* zp = z1    + (size_t)pos[row]   * NUM_COMM;
  const float* zn = z1    + (size_t)neg[row]   * NUM_COMM;

  v8f cp = {}; v8f cn = {};
  #pragma unroll
  for (int s = 0; s < 8; ++s) {
    int k = 4 * s + 2 * h;
    v2f a  = *(const v2f*)(tp + k);
    v2f bp = *(const v2f*)(zp + k);
    v2f bn = *(const v2f*)(zn + k);
    cp = __builtin_amdgcn_wmma_f32_16x16x4_f32(false, a, false, bp, (short)0, cp, false, false);
    cn = __builtin_amdgcn_wmma_f32_16x16x4_f32(false, a, false, bn, (short)0, cn, false, false);
  }
  int base = tile * 16;
  #pragma unroll
  for (int i = 0; i < 8; ++i) {
    int mm = i + ((lane >= 16) ? 8 : 0);
    if ((lane & 15) == mm) {
      gpos[base + mm] = cp[i];
      gneg[base + mm] = cn[i];
    }
  }
}

// ---------------- final BCE reduction -> scalar loss ----------------
__device__ float block_reduce(float v, float* red) {
  int tid = threadIdx.x;
  red[tid] = v; __syncthreads();
  for (int off = blockDim.x >> 1; off > 0; off >>= 1) {
    if (tid < off) red[tid] += red[tid + off];
    __syncthreads();
  }
  float r = red[0]; __syncthreads();
  return r;
}

__global__ void k_final(const float* __restrict__ pos_s, const float* __restrict__ neg_s,
                        const float* __restrict__ gpos, const float* __restrict__ gneg,
                        const float* __restrict__ acc, float* __restrict__ out) {
  __shared__ float red[1024];
  int tid = threadIdx.x;
  float s_mf = 0.0f, s_un = 0.0f, s_gu = 0.0f;
  const float lu1 = -logf(EPS_C);          // bce(EPS, label=1)
  const float lu0 = -logf(1.0f - EPS_C);   // bce(EPS, label=0)
  for (int b = tid; b < BATCH; b += blockDim.x) {
    float pp = 1.0f / (1.0f + expf(-pos_s[b]));
    float pn = 1.0f / (1.0f + expf(-neg_s[b]));
    float gp = gpos[b], gn = gneg[b];
    float bce_p = -fmaxf(logf(pp), LOG_CLAMP);          // label 1
    float bce_n = -fmaxf(logf(1.0f - pn), LOG_CLAMP);   // label 0
    s_mf += gp * bce_p + gn * bce_n;
    s_un += (1.0f - gp) * lu1 + (1.0f - gn) * lu0;
    s_gu += -(gp * fmaxf(logf(gp), LOG_CLAMP) + (1.0f - gp) * fmaxf(logf(1.0f - gp), LOG_CLAMP));
    s_gu += -(gn * fmaxf(logf(gn), LOG_CLAMP) + (1.0f - gn) * fmaxf(logf(1.0f - gn), LOG_CLAMP));
  }
  float t_mf = block_reduce(s_mf, red);
  float t_un = block_reduce(s_un, red);
  float t_gu = block_reduce(s_gu, red);
  if (tid == 0) {
    float inv2B = 1.0f / (2.0f * (float)BATCH);
    float reg1 = 0.5f * acc[0] / (float)BATCH;
    float reg2 = 0.5f * acc[1] / (float)NUM_ITEMS;
    float reg3 = 0.5f * acc[2] / (float)NUM_USERS;
    float regular = WEIGHT_DECAY * (reg1 + reg3) + 0.1f * reg2;
    out[0] = t_mf * inv2B + 0.1f * (t_un * inv2B - t_gu * inv2B) + regular;
  }
}

extern "C" void kernel_launch(void* const* d_in, const int* in_sizes, int n_in,
                              void* d_out, int out_size, void* d_ws, size_t ws_size,
                              hipStream_t stream) {
  const float* ue    = (const float*)d_in[0];
  const float* ie    = (const float*)d_in[1];
  const float* thu   = (const float*)d_in[2];
  const float* w1    = (const float*)d_in[3];
  const float* w2    = (const float*)d_in[4];
  const float* evals = (const float*)d_in[5];
  const int* users   = (const int*)d_in[6];
  const int* pitems  = (const int*)d_in[7];
  const int* nitems  = (const int*)d_in[8];
  const int* erows   = (const int*)d_in[9];
  const int* ecols   = (const int*)d_in[10];

  float* ws    = (float*)d_ws;
  float* theta = ws;                                       // U*C
  float* z     = theta + (size_t)NUM_USERS * NUM_COMM;     // I*C (becomes z1 in place)
  float* pos_s = z + (size_t)NUM_ITEMS * NUM_COMM;         // B
  float* neg_s = pos_s + BATCH;                            // B
  float* gpos  = neg_s + BATCH;                            // B
  float* gneg  = gpos + BATCH;                             // B
  float* acc   = gneg + BATCH;                             // 8 scalars

  k_zero<<<2048, 256, 0, stream>>>(z, NUM_ITEMS * NUM_COMM);
  k_zero<<<1, 32, 0, stream>>>(acc, 8);
  k_softmax<<<1024, 256, 0, stream>>>(thu, theta, acc);
  k_spmm<<<4096, 256, 0, stream>>>(erows, ecols, evals, theta, z);
  k_z1<<<2048, 256, 0, stream>>>(z, w1, w2, acc);
  k_scores<<<BATCH / 16 / 8, 256, 0, stream>>>(ue, ie, users, pitems, nitems, pos_s, neg_s, acc);
  k_gamma<<<BATCH / 16 / 8, 256, 0, stream>>>(theta, z, users, pitems, nitems, gpos, gneg);
  k_final<<<1, 1024, 0, stream>>>(pos_s, neg_s, gpos, gneg, acc, (float*)d_out);
}